// GraphLayer_54065048322680
// MI455X (gfx1250) — compile-verified
//
#include <hip/hip_runtime.h>

// MI455X / gfx1250 fused kernel:
//   h_out = per-field W_out GEMM  ->  aggr = per-batch g GEMM (in-place in LDS)
//   ->  a = per-field W_in GEMM + bias
// All intermediates stay in LDS; HBM traffic = g + h + out ~ 584 MB (roofline ~25us).
// fp32 WMMA (V_WMMA_F32_16X16X4_F32) keeps full reference precision; the op is
// bandwidth-bound so there is no win from dropping precision.

typedef __attribute__((ext_vector_type(2))) float v2f;  // f32 A/B fragment (16x4 / 4x16)
typedef __attribute__((ext_vector_type(8))) float v8f;  // f32 C/D fragment (16x16)

#define BTOT 16384
#define F 50
#define E 64
#define NB 16              // batches per workgroup (one 16-row WMMA M-tile)
#define SF 84              // LDS field stride (floats); 84%64=20 -> half-wave rows land on disjoint banks
#define SB 4204            // LDS batch stride = 50*84+4; 4204%64=44=4*11 -> batch-strided reads conflict-free
#define LDS_BYTES (NB * SB * 4)

static_assert(LDS_BYTES <= 320 * 1024, "exceeds WGP LDS");

__device__ __forceinline__ v8f wmma_f32(v2f a, v2f b, v8f c) {
  // D = A(16x4) * B(4x16) + C, full fp32
  return __builtin_amdgcn_wmma_f32_16x16x4_f32(false, a, false, b, (short)0, c, false, false);
}

extern "C" __global__ __launch_bounds__(256, 1)
void graphlayer_fused(const float* __restrict__ g,
                      const float* __restrict__ h,
                      const float* __restrict__ Wout,
                      const float* __restrict__ Win,
                      const float* __restrict__ bias,
                      float* __restrict__ out)
{
  extern __shared__ float lds[];          // [NB][F][E] padded: b*SB + f*SF + e
  const int tid  = threadIdx.x;
  const int wave = tid >> 5;
  const int lane = tid & 31;
  const int l15  = lane & 15;             // A: row m   | B: col n   (per ISA fragment tables)
  const int hi   = lane >> 4;             // K-pair select: 0 -> K{0,1}, 1 -> K{2,3}
  const int kb   = hi << 1;
  const int hi8  = hi << 3;               // C/D: VGPR v holds row v + 8*hi
  const int b0   = blockIdx.x * NB;

  // Prefetch this tile's g[] (read exactly once, scattered, in phase 2).
  {
    const float* gt = g + (size_t)b0 * (F * F);
    for (int i = tid; i < (NB * F * F) / 32; i += 256)
      __builtin_prefetch(gt + (size_t)i * 32, 0, 1);   // global_prefetch_b8, 128B lines
  }

  // ---------- Phase 1: h_out[b,f,i] = sum_j W_out[f,i,j] h[b,f,j]  -> LDS ----------
  // GEMM per field: M=16 batches, N=64 (4 tiles), K=64. A (h) loaded once, reused 4x.
  #pragma unroll 1
  for (int f = wave; f < F; f += 8) {
    const float* ha = h + ((size_t)(b0 + l15) * F + f) * E + kb;          // A: h[b0+m][f][k..]
    const float* wb = Wout + (size_t)f * (E * E) + l15 * E + kb;          // B: Wout[f][n][k..] (=B[k][n])
    v8f acc[4];
    #pragma unroll
    for (int nt = 0; nt < 4; ++nt) { v8f z = {}; acc[nt] = z; }
    #pragma unroll
    for (int k0 = 0; k0 < E; k0 += 4) {
      const v2f a = *(const v2f*)(ha + k0);
      #pragma unroll
      for (int nt = 0; nt < 4; ++nt) {
        const v2f b = *(const v2f*)(wb + nt * (16 * E) + k0);
        acc[nt] = wmma_f32(a, b, acc[nt]);
      }
    }
    float* dst = lds + f * SF + l15;
    #pragma unroll
    for (int nt = 0; nt < 4; ++nt)
      #pragma unroll
      for (int v = 0; v < 8; ++v)
        dst[(v + hi8) * SB + (nt << 4)] = acc[nt][v];
  }
  __syncthreads();

  // ---------- Phase 2: aggr[b] = g[b](50x50) @ h_out[b](50x64), in place in LDS ----------
  // One wave per batch (2 sequential); 64x64-padded output lives in 128 accumulator VGPRs.
  #pragma unroll 1
  for (int bi = 0; bi < 2; ++bi) {
    const int bb = (wave << 1) | bi;
    const float* gb = g + (size_t)(b0 + bb) * (F * F);
    float* sb = lds + bb * SB;

    v8f acc[4][4];
    #pragma unroll
    for (int mt = 0; mt < 4; ++mt)
      #pragma unroll
      for (int nt = 0; nt < 4; ++nt) { v8f z = {}; acc[mt][nt] = z; }

    // Main K loop: j0+1 <= 47 < F, no guards needed.
    #pragma unroll 1
    for (int k0 = 0; k0 < 48; k0 += 4) {
      const int j0 = k0 + kb;
      v2f bf[4], af[4];
      #pragma unroll
      for (int nt = 0; nt < 4; ++nt) {
        const float* p = sb + j0 * SF + (nt << 4) + l15;
        v2f t; t.x = p[0]; t.y = p[SF];
        bf[nt] = t;
      }
      #pragma unroll
      for (int mt = 0; mt < 4; ++mt) {
        const int m = (mt << 4) + l15;                 // output-field row (pad 50->64)
        v2f t = {};
        if (m < F) t = *(const v2f*)(gb + m * F + j0);
        af[mt] = t;
      }
      #pragma unroll
      for (int mt = 0; mt < 4; ++mt)
        #pragma unroll
        for (int nt = 0; nt < 4; ++nt)
          acc[mt][nt] = wmma_f32(af[mt], bf[nt], acc[mt][nt]);
    }
    // K tail: k = 48..51 (j >= 50 contributes zero)
    {
      const int j0 = 48 + kb;
      v2f bf[4], af[4];
      #pragma unroll
      for (int nt = 0; nt < 4; ++nt) {
        const float* p = sb + j0 * SF + (nt << 4) + l15;
        v2f t = {};
        if (j0 < F)     t.x = p[0];
        if (j0 + 1 < F) t.y = p[SF];
        bf[nt] = t;
      }
      #pragma unroll
      for (int mt = 0; mt < 4; ++mt) {
        const int m = (mt << 4) + l15;
        v2f t = {};
        if (m < F && j0 < F)     t.x = gb[m * F + j0];
        if (m < F && j0 + 1 < F) t.y = gb[m * F + j0 + 1];
        af[mt] = t;
      }
      #pragma unroll
      for (int mt = 0; mt < 4; ++mt)
        #pragma unroll
        for (int nt = 0; nt < 4; ++nt)
          acc[mt][nt] = wmma_f32(af[mt], bf[nt], acc[mt][nt]);
    }
    // Write aggr back over h_out rows of this batch (wave-local; LDS ops are in-order per wave).
    #pragma unroll
    for (int mt = 0; mt < 4; ++mt) {
      #pragma unroll
      for (int v = 0; v < 8; ++v) {
        const int m = (mt << 4) + hi8 + v;
        if (m < F) {
          #pragma unroll
          for (int nt = 0; nt < 4; ++nt)
            sb[m * SF + (nt << 4) + l15] = acc[mt][nt][v];
        }
      }
    }
  }
  __syncthreads();

  // ---------- Phase 3: a[b,f,i] = sum_j W_in[f,i,j] aggr[b,f,j] + bias[i]  -> HBM ----------
  #pragma unroll 1
  for (int f = wave; f < F; f += 8) {
    const float* aa = lds + l15 * SB + f * SF + kb;                       // A: aggr[b=m][f][k..]
    const float* wb = Win + (size_t)f * (E * E) + l15 * E + kb;           // B: Win[f][n][k..]
    v8f acc[4];
    #pragma unroll
    for (int nt = 0; nt < 4; ++nt) {
      const float bv = bias[(nt << 4) + l15];
      v8f t = {bv, bv, bv, bv, bv, bv, bv, bv};
      acc[nt] = t;
    }
    #pragma unroll
    for (int k0 = 0; k0 < E; k0 += 4) {
      const v2f a = *(const v2f*)(aa + k0);
      #pragma unroll
      for (int nt = 0; nt < 4; ++nt) {
        const v2f b = *(const v2f*)(wb + nt * (16 * E) + k0);
        acc[nt] = wmma_f32(a, b, acc[nt]);
      }
    }
    #pragma unroll
    for (int nt = 0; nt < 4; ++nt)
      #pragma unroll
      for (int v = 0; v < 8; ++v)
        out[((size_t)(b0 + hi8 + v) * F + f) * E + (nt << 4) + l15] = acc[nt][v];
  }
}

extern "C" void kernel_launch(void* const* d_in, const int* in_sizes, int n_in,
                              void* d_out, int out_size, void* d_ws, size_t ws_size,
                              hipStream_t stream) {
  (void)in_sizes; (void)n_in; (void)out_size; (void)d_ws; (void)ws_size;
  const float* g    = (const float*)d_in[0];
  const float* h    = (const float*)d_in[1];
  const float* Win  = (const float*)d_in[2];
  const float* Wout = (const float*)d_in[3];
  const float* bias = (const float*)d_in[4];
  float* out = (float*)d_out;

  // 269 KB dynamic LDS per workgroup (<= 320 KB WGP budget); opt in every call
  // (deterministic, not a stream op -> graph-capture safe).
  (void)hipFuncSetAttribute((const void*)graphlayer_fused,
                            hipFuncAttributeMaxDynamicSharedMemorySize, LDS_BYTES);

  graphlayer_fused<<<BTOT / NB, 256, LDS_BYTES, stream>>>(g, h, Wout, Win, bias, out);
}